// JaggedConv2D_41798621724854
// MI455X (gfx1250) — compile-verified
//
#include <hip/hip_runtime.h>

#define B_   4
#define C_   128
#define H_   256
#define W_   256
#define KM   29
#define RAD  14
#define TILE 64
#define HALO (TILE + 2 * RAD)   /* 92 */
#define ST   96                 /* padded LDS row stride in floats (16B aligned) */
#define NTHR 256

typedef float v2f __attribute__((ext_vector_type(2)));

#if defined(__HIP_DEVICE_COMPILE__) && __has_builtin(__builtin_amdgcn_global_load_async_to_lds_b32)
#define HAVE_ASYNC_LDS 1
#else
#define HAVE_ASYNC_LDS 0
#endif

#if defined(__HIP_DEVICE_COMPILE__) && __has_builtin(__builtin_amdgcn_s_wait_asynccnt)
#define HAVE_WAIT_ASYNC 1
#else
#define HAVE_WAIT_ASYNC 0
#endif

#if HAVE_ASYNC_LDS
typedef __attribute__((address_space(1))) int g_i32;   // global (AS1)
typedef __attribute__((address_space(3))) int l_i32;   // LDS (AS3)
#endif

// Force packed fp32 FMA (VOP3P V_PK_FMA_F32: D = S0*S1 + S2, 2 x f32 per issue).
// The compiler scalarized <2 x float> fma, so emit the instruction directly.
#if defined(__HIP_DEVICE_COMPILE__) && defined(__gfx1250__)
#define PK_FMA(acc, a, b) \
    asm("v_pk_fma_f32 %0, %1, %2, %0" : "+v"(acc) : "v"(a), "v"(b))
#else
#define PK_FMA(acc, a, b) ((acc) = __builtin_elementwise_fma((a), (b), (acc)))
#endif

__global__ __launch_bounds__(NTHR)
void JaggedConv2D_kernel(const float* __restrict__ xin,
                         const float* __restrict__ kern,
                         float* __restrict__ out)
{
    __shared__ float sIn[HALO * ST];   // 35,328 B
    __shared__ v2f   sW2[KM * KM];     //  6,728 B (weights pre-duplicated {w,w})

    const int tid    = threadIdx.x;
    const int ch     = blockIdx.y;
    const int b      = blockIdx.z;
    const int tileX0 = (blockIdx.x & 3) * TILE;
    const int tileY0 = (blockIdx.x >> 2) * TILE;

    const float* __restrict__ gin = xin + (size_t)(b * C_ + ch) * (H_ * W_);

    // Ragged kernel extent: reference uses k = floor(5 + 24*ch/127) forced odd
    // DOWNWARD. We force odd UPWARD -> our centered window is always a superset;
    // the extra taps hit stored zeros, so this is safe and skips ~60% of taps
    // for small-k channels.
    int kf = (int)(5.0 + (24.0 * (double)ch) / 127.0 + 1e-9);
    kf += 1 - (kf & 1);
    if (kf > KM) kf = KM;
    const int s = (KM - kf) >> 1;

    // Stage this channel's 29x29 weights into LDS, duplicated into both halves
    // of a 64-bit pair so the inner loop gets {w,w} with one ds_load_b64
    // broadcast (no per-tap splat mov).
    for (int i = tid; i < KM * KM; i += NTHR) {
        const float w = kern[ch * (KM * KM) + i];
        sW2[i] = (v2f){w, w};
    }

    // Stage zero-padded input halo into LDS. In-bounds lanes use the CDNA5
    // async global->LDS path (ASYNCcnt); out-of-bounds slots are zero-filled
    // with plain ds stores (disjoint addresses, no ordering hazard).
    for (int i = tid; i < HALO * HALO; i += NTHR) {
        const int r  = i / HALO;
        const int c  = i - r * HALO;
        const int gy = tileY0 + r - RAD;
        const int gx = tileX0 + c - RAD;
        const int lo = r * ST + c;
        if ((unsigned)gy < (unsigned)H_ && (unsigned)gx < (unsigned)W_) {
#if HAVE_ASYNC_LDS
            __builtin_amdgcn_global_load_async_to_lds_b32(
                (g_i32*)(uintptr_t)(gin + gy * W_ + gx),
                (l_i32*)&sIn[lo],
                0, 0);
#else
            sIn[lo] = gin[gy * W_ + gx];
#endif
        } else {
            sIn[lo] = 0.0f;
        }
    }

#if HAVE_ASYNC_LDS && HAVE_WAIT_ASYNC
    __builtin_amdgcn_s_wait_asynccnt(0);
#endif
    __syncthreads();

    // 16x16 threads, each owns a 4x4 output tile (4 consecutive x, 4 consecutive y).
    // Accumulators are packed across row pairs {0,1} and {2,3}: same weight
    // multiplies both halves -> v_pk_fma_f32 (2 fp32 FMAs per VALU issue).
    const int tix = tid & 15;
    const int tiy = tid >> 4;
    const int x0  = tix * 4;
    const int y0  = tiy * 4;

    v2f accA0 = {0.f, 0.f}, accA1 = {0.f, 0.f}, accA2 = {0.f, 0.f}, accA3 = {0.f, 0.f};
    v2f accB0 = {0.f, 0.f}, accB1 = {0.f, 0.f}, accB2 = {0.f, 0.f}, accB3 = {0.f, 0.f};

    for (int dy = s; dy < s + kf; ++dy) {
        const v2f*   __restrict__ wr = &sW2[dy * KM + s];
        const float* __restrict__ p0 = &sIn[(y0 + dy) * ST + x0 + s];
        const float* __restrict__ p1 = p0 + ST;
        const float* __restrict__ p2 = p1 + ST;
        const float* __restrict__ p3 = p2 + ST;

        // sliding 4-wide window per row pair: per tap, 4 new ds_load_b32 (into
        // 64-bit pair halves) + 1 ds_load_b64 weight broadcast feed 8 pk-FMAs.
        v2f A0 = {p0[0], p1[0]}, A1 = {p0[1], p1[1]}, A2 = {p0[2], p1[2]};
        v2f B0 = {p2[0], p3[0]}, B1 = {p2[1], p3[1]}, B2 = {p2[2], p3[2]};

#pragma unroll 3
        for (int t = 0; t < kf; ++t) {
            const v2f wv2 = wr[t];
            v2f A3 = {p0[t + 3], p1[t + 3]};
            v2f B3 = {p2[t + 3], p3[t + 3]};

            PK_FMA(accA0, wv2, A0);
            PK_FMA(accA1, wv2, A1);
            PK_FMA(accA2, wv2, A2);
            PK_FMA(accA3, wv2, A3);
            PK_FMA(accB0, wv2, B0);
            PK_FMA(accB1, wv2, B1);
            PK_FMA(accB2, wv2, B2);
            PK_FMA(accB3, wv2, B3);

            A0 = A1; A1 = A2; A2 = A3;
            B0 = B1; B1 = B2; B2 = B3;
        }
    }

    float* gout = out + ((size_t)(b * C_ + ch) * H_ + (tileY0 + y0)) * W_
                      + tileX0 + x0;
    *(float4*)(gout + 0 * W_) = make_float4(accA0.x, accA1.x, accA2.x, accA3.x);
    *(float4*)(gout + 1 * W_) = make_float4(accA0.y, accA1.y, accA2.y, accA3.y);
    *(float4*)(gout + 2 * W_) = make_float4(accB0.x, accB1.x, accB2.x, accB3.x);
    *(float4*)(gout + 3 * W_) = make_float4(accB0.y, accB1.y, accB2.y, accB3.y);
}

extern "C" void kernel_launch(void* const* d_in, const int* in_sizes, int n_in,
                              void* d_out, int out_size, void* d_ws, size_t ws_size,
                              hipStream_t stream)
{
    (void)in_sizes; (void)n_in; (void)out_size; (void)d_ws; (void)ws_size;
    const float* x    = (const float*)d_in[0];   // (4,128,256,256) f32
    const float* kern = (const float*)d_in[1];   // (128,1,29,29)   f32
    float*       out  = (float*)d_out;           // (4,128,256,256) f32

    dim3 grid((H_ / TILE) * (W_ / TILE), C_, B_); // 16 x 128 x 4
    dim3 block(NTHR);
    hipLaunchKernelGGL(JaggedConv2D_kernel, grid, block, 0, stream, x, kern, out);
}